// OPIQNoveltyModule_48945447305478
// MI455X (gfx1250) — compile-verified
//
#include <hip/hip_runtime.h>

// ---------------------------------------------------------------------------
// OPIQ novelty: S = A @ [states|actions]^T (+b), sign -> bitpack-hash -> gather
//   B=2048 (batch == hash_size), D=2047, K=4096, TABLE=1<<22
// FP32 WMMA (v_wmma_f32_16x16x4_f32) GEMM (compute bound: ~690 FLOP/byte;
// sign() semantics require fp32 accumulation). Staging uses the CDNA5
// GLOBAL_LOAD_ASYNC_TO_LDS_B128 DMA path with double-buffered LDS and
// s_wait_asynccnt, overlapping DMA with the WMMA chains.
// Hash accumulated with wrap-around u32 atomics: (x mod 2^32) & (2^22-1) is
// exact because 2^22 divides 2^32.
// ---------------------------------------------------------------------------

typedef __attribute__((ext_vector_type(2))) float v2f;
typedef __attribute__((ext_vector_type(8))) float v8f;

#define B_DIM 2048
#define D_DIM 2047
#define K_DIM 4096
#define HASH  2048              // D+1, GEMM contraction length
#define TABLE_MASK 0x3FFFFFu    // (1<<22)-1

#define BM 128                  // block tile along K_DIM (rows of A)
#define BN 32                   // block tile along batch (2 accum per wave)
#define KC 32                   // K-chunk staged in LDS
#define LDS_STRIDE 36           // 144B row stride: 16B-aligned rows for B128
                                // LDS writes; 36*m mod 64 -> 16 distinct banks

// CDNA5 async DMA: global -> LDS, 16 bytes per lane, tracked by ASYNCcnt.
// vdst = 32-bit LDS byte address (flat LDS addr truncates to [31:0]),
// vaddr = 64-bit global address pair.
__device__ __forceinline__ void async_copy_b128(const float* gp, float* lp) {
    unsigned           lds = (unsigned)(size_t)lp;
    unsigned long long ga  = (unsigned long long)(size_t)gp;
    asm volatile("global_load_async_to_lds_b128 %0, %1, off"
                 :: "v"(lds), "v"(ga) : "memory");
}
__device__ __forceinline__ void wait_async0() {
    asm volatile("s_wait_asynccnt 0x0" ::: "memory");
}

// ---------------------------------------------------------------------------
// Kernel 1: pack W[i][d] = (d < 2047) ? states[i][d] : actions[i];  zero acc.
// ---------------------------------------------------------------------------
__global__ void pack_kernel(const float* __restrict__ states,
                            const float* __restrict__ actions,
                            float* __restrict__ W,
                            unsigned* __restrict__ acc) {
    int idx = blockIdx.x * blockDim.x + threadIdx.x;   // 0 .. 2048*2048-1
    int i = idx >> 11;
    int d = idx & 2047;
    W[idx] = (d < D_DIM) ? states[(size_t)i * D_DIM + d] : actions[i];
    if (idx < B_DIM) acc[idx] = 0u;
}

// ---------------------------------------------------------------------------
// Kernel 2: double-buffered async-DMA FP32 WMMA GEMM + sign + hash partials.
// Grid (K_DIM/BM, B_DIM/BN); 256 threads = 8 waves; wave w owns two 16x16
// tiles: rows [m0+16w, +16) x cols [n0, +16) and [n0+16, +32).
// ---------------------------------------------------------------------------
__global__ __launch_bounds__(256)
void gemm_sign_hash_kernel(const float* __restrict__ A,      // [K_DIM][HASH]
                           const float* __restrict__ W,      // [B_DIM][HASH]
                           const float* __restrict__ bvec,   // [B_DIM]
                           const int*   __restrict__ coeffs, // [K_DIM]
                           unsigned*    __restrict__ acc) {  // [B_DIM]
    __shared__ float At[2][BM][LDS_STRIDE];
    __shared__ float Bt[2][BN][LDS_STRIDE];

    const int m0   = blockIdx.x * BM;
    const int n0   = blockIdx.y * BN;
    const int t    = threadIdx.x;
    const int wave = t >> 5;          // 0..7 (wave32)
    const int lane = t & 31;
    const int lrow = lane & 15;       // M-row (A frag) / N-col (B frag)
    const int half = lane >> 4;       // selects K pair {0,1} vs {2,3}

    // async-issue one K-chunk into LDS buffer `buf`
    auto issue_chunk = [&](int buf, int kc) {
        // A tile: 128 rows x 32 floats = 1024 x b128, 4 per thread
        #pragma unroll
        for (int it = 0; it < 4; ++it) {
            int idx = t + 256 * it;
            int row = idx >> 3;
            int seg = idx & 7;
            async_copy_b128(&A[(size_t)(m0 + row) * HASH + kc + seg * 4],
                            &At[buf][row][seg * 4]);
        }
        // B tile: 32 rows x 32 floats = 256 x b128, 1 per thread
        {
            int row = t >> 3;
            int seg = t & 7;
            async_copy_b128(&W[(size_t)(n0 + row) * HASH + kc + seg * 4],
                            &Bt[buf][row][seg * 4]);
        }
    };

    v8f c0 = {0.f,0.f,0.f,0.f,0.f,0.f,0.f,0.f};
    v8f c1 = {0.f,0.f,0.f,0.f,0.f,0.f,0.f,0.f};

    // prologue: DMA chunk 0
    issue_chunk(0, 0);
    wait_async0();
    __syncthreads();

    const int arow = 16 * wave + lrow;
    int buf = 0;
    for (int kc = 0; kc < HASH; kc += KC, buf ^= 1) {
        // overlap: DMA next chunk into the other buffer while we compute
        if (kc + KC < HASH) issue_chunk(buf ^ 1, kc + KC);

        // 16 x v_wmma_f32_16x16x4_f32 per chunk (2 independent chains)
        #pragma unroll
        for (int kk = 0; kk < KC; kk += 4) {
            // A frag: lane holds A[m = lane&15, k = kk + 2*half + {0,1}]
            v2f a  = *(const v2f*)&At[buf][arow][kk + 2 * half];
            // B frags: lane holds Bmat[k, n = lane&15] = W[n0(+16) + n][k]
            v2f b0 = *(const v2f*)&Bt[buf][lrow][kk + 2 * half];
            v2f b1 = *(const v2f*)&Bt[buf][16 + lrow][kk + 2 * half];
            c0 = __builtin_amdgcn_wmma_f32_16x16x4_f32(
                     false, a, false, b0, (short)0, c0, false, false);
            c1 = __builtin_amdgcn_wmma_f32_16x16x4_f32(
                     false, a, false, b1, (short)0, c1, false, false);
        }

        wait_async0();     // our next-chunk DMA has landed in LDS
        __syncthreads();   // all waves done reading buf / all DMAs visible
    }

    // --- epilogue: +b[i], sign, per-column coefficient sum (u32 wraparound)
    const int   nc0 = n0 + lrow;
    const int   nc1 = n0 + 16 + lrow;
    const float bb0 = bvec[nc0];
    const float bb1 = bvec[nc1];
    unsigned h0 = 0, h1 = 0;
    #pragma unroll
    for (int r = 0; r < 8; ++r) {
        // C/D layout: VGPR r holds M = r + 8*half, N = lane&15
        unsigned w = (unsigned)coeffs[m0 + 16 * wave + r + 8 * half];
        if (c0[r] + bb0 > 0.0f) h0 += w;
        if (c1[r] + bb1 > 0.0f) h1 += w;
    }
    // fold lanes 16..31 (M rows 8..15) into lanes 0..15 (same column)
    h0 += __shfl_xor(h0, 16, 32);
    h1 += __shfl_xor(h1, 16, 32);
    if (half == 0) {
        atomicAdd(&acc[nc0], h0);
        atomicAdd(&acc[nc1], h1);
    }
}

// ---------------------------------------------------------------------------
// Kernel 3: idx = acc & (2^22-1); gather; novelty = 1/(count+1)^2
// ---------------------------------------------------------------------------
__global__ void finalize_kernel(const unsigned* __restrict__ acc,
                                const int* __restrict__ table,
                                float* __restrict__ out) {
    int i = blockIdx.x * blockDim.x + threadIdx.x;
    if (i < B_DIM) {
        unsigned idx = acc[i] & TABLE_MASK;
        float cnt = (float)table[idx] + 1.0f;
        out[i] = 1.0f / (cnt * cnt);
    }
}

// ---------------------------------------------------------------------------
extern "C" void kernel_launch(void* const* d_in, const int* in_sizes, int n_in,
                              void* d_out, int out_size, void* d_ws, size_t ws_size,
                              hipStream_t stream) {
    const float* states  = (const float*)d_in[0];  // [2048, 2047]
    const float* actions = (const float*)d_in[1];  // [2048]
    const float* A       = (const float*)d_in[2];  // [4096, 2048]
    const float* bvec    = (const float*)d_in[3];  // [2048]
    const int*   table   = (const int*)d_in[4];    // [1<<22]
    const int*   coeffs  = (const int*)d_in[5];    // [4096]
    float*       out     = (float*)d_out;          // [2048]

    float*    W   = (float*)d_ws;                                      // 16 MB
    unsigned* acc = (unsigned*)((char*)d_ws +
                                (size_t)B_DIM * HASH * sizeof(float)); // +8 KB

    // 1) pack states|actions into W, zero hash accumulators (every call)
    pack_kernel<<<(B_DIM * HASH) / 256, 256, 0, stream>>>(states, actions, W, acc);

    // 2) async-DMA double-buffered WMMA GEMM + sign + hash partials
    dim3 grid(K_DIM / BM, B_DIM / BN);   // 32 x 64 blocks
    gemm_sign_hash_kernel<<<grid, 256, 0, stream>>>(A, W, bvec, coeffs, acc);

    // 3) gather + novelty
    finalize_kernel<<<(B_DIM + 255) / 256, 256, 0, stream>>>(acc, table, out);
}